// Bidir_64510408786371
// MI455X (gfx1250) — compile-verified
//
#include <hip/hip_runtime.h>

// Problem constants (from reference)
#define BB    8
#define LL    512
#define DD    768
#define HH    384          // D/2
#define MAXW  12
#define SS    (LL * MAXW)  // 6144 spans per batch
#define MTOT  (BB * SS)    // 49152 rows

// Tiling: block = 128(M) x 256(N), 8 waves as 2(M) x 4(N), wave tile 64x64
#define BM 128
#define BN 256
#define KC 16

typedef float v2f __attribute__((ext_vector_type(2)));
typedef float v8f __attribute__((ext_vector_type(8)));

__device__ __forceinline__ float4 ldg4(const float* p) {
    return *(const float4*)p;
}

__global__ __launch_bounds__(256)
void span_gemm_wmma(const float* __restrict__ h,
                    const int*   __restrict__ span_idx,
                    const float* __restrict__ W,
                    const float* __restrict__ bias,
                    const float* __restrict__ ssent,
                    const float* __restrict__ esent,
                    float*       __restrict__ out)
{
    // k-pair layouts: element (k, r) lives at T[k>>1][r][k&1]; one fragment = one
    // aligned 8-byte v2f -> single ds_load_b64, conflict-free (+16 pad => b64-slot
    // stride per k-pair is 16 mod 32, separating the two lane halves).
    __shared__ v2f  A2[KC / 2][BM + 16];  // gathered span_rep tile
    __shared__ v2f  B2[KC / 2][BN + 16];  // W^T tile: (k,n) = W[nTile+n][kbase+k]
    __shared__ int  rowOff[BM][4];        // gather row offsets (floats); -1 => sentinel
    __shared__ float sent[DD];            // [0..383]=start_sentinel, [384..767]=end_sentinel

    const int tid   = threadIdx.x;
    const int nTile = blockIdx.x * BN;
    const int mTile = blockIdx.y * BM;

    // Cache sentinels in LDS (indexed directly by global column d)
    for (int i = tid; i < DD; i += 256)
        sent[i] = (i < HH) ? ssent[i] : esent[i - HH];

    // Per-row gather descriptors
    if (tid < BM) {
        int g     = mTile + tid;
        int b     = g / SS;
        int s     = g - b * SS;
        int start = span_idx[(b * SS + s) * 2 + 0];
        int end   = span_idx[(b * SS + s) * 2 + 1];
        int base  = b * LL;
        rowOff[tid][0] = (base + end) * DD;                              // f_end row
        rowOff[tid][1] = (start == 0)    ? -1 : (base + start - 1) * DD; // f_start / start_sentinel
        rowOff[tid][2] = (end >= LL - 1) ? -1 : (base + end + 1) * DD;   // b_start / end_sentinel
        rowOff[tid][3] = (base + start) * DD;                            // b_end row
    }

    const int lane     = tid & 31;
    const int wave     = tid >> 5;
    const int wm       = (wave & 1) << 6;   // wave M offset: 0/64
    const int wn       = (wave >> 1) << 6;  // wave N offset: 0/64/128/192
    const int laneLo   = lane & 15;
    const int laneHalf = lane >> 4;

    v8f acc[4][4];
    #pragma unroll
    for (int i = 0; i < 4; ++i)
        #pragma unroll
        for (int j = 0; j < 4; ++j)
            acc[i][j] = (v8f){0.f, 0.f, 0.f, 0.f, 0.f, 0.f, 0.f, 0.f};

    // A staging: thread t owns row r = t&127, column half dp = (t>>7)*8
    const int ar  = tid & 127;
    const int adp = (tid >> 7) * 8;

    for (int kbase = 0; kbase < DD; kbase += KC) {
        __syncthreads();   // LDS reuse guard

        // ---- Stage A tile: gathered span_rep[mTile..+127][kbase..+15] ----
        {
            int d0 = kbase + adp;
            const int* ro = rowOff[ar];
            float v[8];
            if (kbase < HH) {                       // forward half: f_end - f_start
                float4 p0 = ldg4(h + ro[0] + d0);
                float4 p1 = ldg4(h + ro[0] + d0 + 4);
                v[0]=p0.x; v[1]=p0.y; v[2]=p0.z; v[3]=p0.w;
                v[4]=p1.x; v[5]=p1.y; v[6]=p1.z; v[7]=p1.w;
                if (ro[1] < 0) {
                    #pragma unroll
                    for (int i = 0; i < 8; ++i) v[i] -= sent[d0 + i];
                } else {
                    float4 q0 = ldg4(h + ro[1] + d0);
                    float4 q1 = ldg4(h + ro[1] + d0 + 4);
                    v[0]-=q0.x; v[1]-=q0.y; v[2]-=q0.z; v[3]-=q0.w;
                    v[4]-=q1.x; v[5]-=q1.y; v[6]-=q1.z; v[7]-=q1.w;
                }
            } else {                                // backward half: b_start - b_end
                if (ro[2] < 0) {
                    #pragma unroll
                    for (int i = 0; i < 8; ++i) v[i] = sent[d0 + i];  // d0>=384 -> end_sentinel
                } else {
                    float4 p0 = ldg4(h + ro[2] + d0);
                    float4 p1 = ldg4(h + ro[2] + d0 + 4);
                    v[0]=p0.x; v[1]=p0.y; v[2]=p0.z; v[3]=p0.w;
                    v[4]=p1.x; v[5]=p1.y; v[6]=p1.z; v[7]=p1.w;
                }
                float4 q0 = ldg4(h + ro[3] + d0);
                float4 q1 = ldg4(h + ro[3] + d0 + 4);
                v[0]-=q0.x; v[1]-=q0.y; v[2]-=q0.z; v[3]-=q0.w;
                v[4]-=q1.x; v[5]-=q1.y; v[6]-=q1.z; v[7]-=q1.w;
            }
            #pragma unroll
            for (int i = 0; i < 4; ++i)
                A2[(adp >> 1) + i][ar] = (v2f){v[2 * i], v[2 * i + 1]};
        }

        // ---- Stage B tile: W rows -> k-pair transposed LDS ----
        {
            const float* wr = W + (size_t)(nTile + tid) * DD + kbase;
            float4 w0 = ldg4(wr);
            float4 w1 = ldg4(wr + 4);
            float4 w2 = ldg4(wr + 8);
            float4 w3 = ldg4(wr + 12);
            B2[0][tid] = (v2f){w0.x, w0.y};
            B2[1][tid] = (v2f){w0.z, w0.w};
            B2[2][tid] = (v2f){w1.x, w1.y};
            B2[3][tid] = (v2f){w1.z, w1.w};
            B2[4][tid] = (v2f){w2.x, w2.y};
            B2[5][tid] = (v2f){w2.z, w2.w};
            B2[6][tid] = (v2f){w3.x, w3.y};
            B2[7][tid] = (v2f){w3.z, w3.w};
        }

        __syncthreads();

        // ---- Compute: 4 K-steps x 16 V_WMMA_F32_16X16X4_F32 ----
        #pragma unroll
        for (int k0 = 0; k0 < KC; k0 += 4) {
            // ISA 7.12.2: operand VGPR v, lane-half hh holds K = k0 + hh*2 + v
            int k2 = (k0 >> 1) + laneHalf;
            v2f afrag[4], bfrag[4];
            #pragma unroll
            for (int i = 0; i < 4; ++i)
                afrag[i] = A2[k2][wm + i * 16 + laneLo];
            #pragma unroll
            for (int j = 0; j < 4; ++j)
                bfrag[j] = B2[k2][wn + j * 16 + laneLo];
            #pragma unroll
            for (int i = 0; i < 4; ++i)
                #pragma unroll
                for (int j = 0; j < 4; ++j)
                    acc[i][j] = __builtin_amdgcn_wmma_f32_16x16x4_f32(
                        /*neg_a=*/false, afrag[i],
                        /*neg_b=*/false, bfrag[j],
                        /*c_mod=*/(short)0, acc[i][j],
                        /*reuse_a=*/false, /*reuse_b=*/false);
        }
    }

    // ---- Epilogue: relu(acc + bias); C/D layout: VGPR v -> rows v and v+8 ----
    #pragma unroll
    for (int j = 0; j < 4; ++j) {
        int col  = nTile + wn + j * 16 + laneLo;
        float bv = bias[col];
        #pragma unroll
        for (int i = 0; i < 4; ++i) {
            #pragma unroll
            for (int v = 0; v < 8; ++v) {
                int row = mTile + wm + i * 16 + v + laneHalf * 8;
                float x = acc[i][j][v] + bv;
                out[(size_t)row * DD + col] = fmaxf(x, 0.f);
            }
        }
    }
}

extern "C" void kernel_launch(void* const* d_in, const int* in_sizes, int n_in,
                              void* d_out, int out_size, void* d_ws, size_t ws_size,
                              hipStream_t stream) {
    (void)in_sizes; (void)n_in; (void)out_size; (void)d_ws; (void)ws_size;
    const float* h        = (const float*)d_in[0];
    const int*   span_idx = (const int*)d_in[1];
    const float* W        = (const float*)d_in[2];
    const float* bias     = (const float*)d_in[3];
    const float* ssent    = (const float*)d_in[4];
    const float* esent    = (const float*)d_in[5];
    float*       out      = (float*)d_out;

    dim3 grid(DD / BN, MTOT / BM);   // 3 x 384
    span_gemm_wmma<<<grid, 256, 0, stream>>>(h, span_idx, W, bias, ssent, esent, out);
}